// GMF_86062554677779
// MI455X (gfx1250) — compile-verified
//
#include <hip/hip_runtime.h>

// GMF triple-scoring on gfx1250 (MI455X, wave32, WMMA).
//
//   out_w[b,i] = sum_d u[b,d] * w[d] * v[i,d],  w in {view, cart, buy}
//
// Strategy:
//   Kernel 1 (gmf_prep): gather user rows and fold each weight vector in:
//       A3[w][b][d] = user_table[batch_users[b]][d] * weight_w[d]   (768 KB scratch)
//   Kernel 2 (gmf_score): three fused fp32 GEMMs A3[w] (1024x64) @ v^T (64x50000)
//       using V_WMMA_F32_16X16X4_F32, sharing one B (item) fragment across the
//       three weight accumulators. Store-bandwidth-bound (614 MB out, ~26us floor).

typedef float v2f __attribute__((ext_vector_type(2)));
typedef float v8f __attribute__((ext_vector_type(8)));

#define GMF_DIM   64
#define GMF_BATCH 1024

__global__ __launch_bounds__(256) void gmf_prep(
    const int*   __restrict__ users,       // [BATCH]
    const float* __restrict__ user_table,  // [NUM_USERS][DIM]
    const float* __restrict__ wv,          // [DIM]
    const float* __restrict__ wc,          // [DIM]
    const float* __restrict__ wb,          // [DIM]
    float*       __restrict__ A3)          // [3][BATCH][DIM]
{
    const int idx = blockIdx.x * blockDim.x + threadIdx.x;
    if (idx >= GMF_BATCH * GMF_DIM) return;
    const int b = idx >> 6;
    const int d = idx & 63;
    const float u = user_table[(size_t)users[b] * GMF_DIM + d];
    A3[idx]                          = u * wv[d];
    A3[GMF_BATCH * GMF_DIM + idx]    = u * wc[d];
    A3[2 * GMF_BATCH * GMF_DIM + idx] = u * wb[d];
}

// Each wave computes one 16x16 output tile for all three weights.
// Block: 256 threads = 8 waves covering 16 rows (M) x 128 cols (N).
__global__ __launch_bounds__(256) void gmf_score(
    const float* __restrict__ A3,          // [3][BATCH][DIM]  (w-scaled user rows)
    const float* __restrict__ item_table,  // [NUM_ITEMS][DIM]
    const int*   __restrict__ items,       // [NI] (identity gather in practice)
    float*       __restrict__ out,         // [3][BATCH][NI]
    int NI)
{
    const int lane  = threadIdx.x & 31;
    const int wave  = threadIdx.x >> 5;            // 0..7
    const int mBase = blockIdx.y << 4;             // 16-row tile of B(atch)
    const int nBase = (blockIdx.x << 7) + (wave << 4);
    const int half  = lane >> 4;                   // 0 or 1
    const int row   = lane & 15;
    const int khalf = half << 1;                   // K sub-offset: 0 or 2

    // A fragment base (16x4 fp32 tile): lane l -> float2 at A[row=l%16][k + 2*(l/16)]
    const float* a0 = A3 + (size_t)(mBase + row) * GMF_DIM + khalf;          // view
    const float* a1 = a0 + (size_t)GMF_BATCH * GMF_DIM;                      // cart
    const float* a2 = a1 + (size_t)GMF_BATCH * GMF_DIM;                      // buy

    // B fragment base (4x16 fp32 tile of v^T): lane l -> float2 at
    // v[item(nBase + l%16)][k + 2*(l/16)]. Clamp the ragged edge so the
    // wave never predicates around the WMMAs (EXEC must stay all-1s).
    int ncl = nBase + row;
    if (ncl >= NI) ncl = NI - 1;
    const int   iidx = items[ncl];
    const float* bp  = item_table + (size_t)iidx * GMF_DIM + khalf;

    v8f accV = {}; v8f accC = {}; v8f accB = {};

#pragma unroll
    for (int k = 0; k < GMF_DIM; k += 4) {
        const v2f bf = *(const v2f*)(bp + k);      // shared item fragment
        const v2f av = *(const v2f*)(a0 + k);
        const v2f ac = *(const v2f*)(a1 + k);
        const v2f ab = *(const v2f*)(a2 + k);
        accV = __builtin_amdgcn_wmma_f32_16x16x4_f32(false, av, false, bf,
                                                     (short)0, accV, false, false);
        accC = __builtin_amdgcn_wmma_f32_16x16x4_f32(false, ac, false, bf,
                                                     (short)0, accC, false, false);
        accB = __builtin_amdgcn_wmma_f32_16x16x4_f32(false, ab, false, bf,
                                                     (short)0, accB, false, false);
    }

    // D layout: VGPR r -> (M = r + 8*half, N = lane%16).
    const int n = nBase + row;
    if (n < NI) {
        const size_t plane = (size_t)GMF_BATCH * NI;
        float* o = out + (size_t)(mBase + (half << 3)) * NI + n;
#pragma unroll
        for (int r = 0; r < 8; ++r) {
            o[(size_t)r * NI]               = accV[r];
            o[plane + (size_t)r * NI]       = accC[r];
            o[2 * plane + (size_t)r * NI]   = accB[r];
        }
    }
}

extern "C" void kernel_launch(void* const* d_in, const int* in_sizes, int n_in,
                              void* d_out, int out_size, void* d_ws, size_t ws_size,
                              hipStream_t stream) {
    // setup_inputs() order:
    //   0: batch_users (int32 [1024])
    //   1: whole_items (int32 [50000])
    //   2: dropout_ration (scalar, identity -> unused)
    //   3: user_table  (f32 [100000,64])
    //   4: item_table  (f32 [50000,64])
    //   5: weight_view (f32 [64,1])
    //   6: weight_cart (f32 [64,1])
    //   7: weight_buy  (f32 [64,1])
    const int*   users = (const int*)  d_in[0];
    const int*   items = (const int*)  d_in[1];
    const float* utab  = (const float*)d_in[3];
    const float* itab  = (const float*)d_in[4];
    const float* wv    = (const float*)d_in[5];
    const float* wc    = (const float*)d_in[6];
    const float* wb    = (const float*)d_in[7];
    float* out = (float*)d_out;
    float* A3  = (float*)d_ws;                 // 3*1024*64*4 = 768 KB scratch

    const int NI = in_sizes[1];

    gmf_prep<<<(GMF_BATCH * GMF_DIM + 255) / 256, 256, 0, stream>>>(
        users, utab, wv, wc, wb, A3);

    dim3 grid((NI + 127) / 128, GMF_BATCH / 16);
    gmf_score<<<grid, 256, 0, stream>>>(A3, itab, items, out, NI);
}